// _RoICrop_42606075576992
// MI455X (gfx1250) — compile-verified
//
#include <hip/hip_runtime.h>
#include <hip/hip_bf16.h>

// Bilinear RoI crop (grid sample), BHWD layout.
//   feat: [B=128, H=37, W=50, C=512] f32
//   grid: [B, GH=14, GW=14, 2] f32 (y,x) in [-1,1]
//   out : [B, GH, GW, C] f32
//
// Memory-bound gather (~0.1 GFLOP vs ~200+ MB traffic -> AI ~0.4 flop/B).
// Strategy: one wave32 per output pixel; wave-uniform scalar addressing
// (readfirstlane -> SADDR global loads); lane-coalesced float4 (B128) gathers
// of the 4 corner rows; FMA blend; non-temporal B128 stores (output is
// write-once, keep L2 for the reusable feature rows).

typedef float v4f __attribute__((ext_vector_type(4)));

__global__ __launch_bounds__(256) void _RoICrop_kernel(
    const float* __restrict__ feat,
    const float* __restrict__ grid,
    float* __restrict__ out)
{
    constexpr int B  = 128;
    constexpr int H  = 37;
    constexpr int W  = 50;
    constexpr int C  = 512;
    constexpr int GH = 14;
    constexpr int GW = 14;
    constexpr int TOTAL = B * GH * GW;           // 25088 output pixels
    constexpr int V4_PER_ROW = C / 4;            // 128 float4 per feature row
    constexpr int ITERS = V4_PER_ROW / 32;       // 4 float4 per lane

    const int lane = threadIdx.x & 31;
    const int waveInBlock = threadIdx.x >> 5;    // 8 waves per 256-thread block
    int pixel = blockIdx.x * (blockDim.x >> 5) + waveInBlock;

    // pixel is uniform across the wave; make it an SGPR so all derived
    // address math lands on the SALU and loads use the SADDR form.
    pixel = __builtin_amdgcn_readfirstlane(pixel);
    if (pixel >= TOTAL) return;

    const int b = pixel / (GH * GW);

    // Grid coords for this pixel (wave-uniform load -> scalar path).
    const float gy = grid[2 * pixel + 0];
    const float gx = grid[2 * pixel + 1];

    // Map [-1,1] -> [0, size-1], exactly as the reference.
    const float yf = (gy + 1.0f) * 0.5f * (float)(H - 1);
    const float xf = (gx + 1.0f) * 0.5f * (float)(W - 1);
    const float y0f = floorf(yf);
    const float x0f = floorf(xf);
    const float wy = yf - y0f;
    const float wx = xf - x0f;

    int y0 = (int)y0f; y0 = y0 < 0 ? 0 : (y0 > H - 1 ? H - 1 : y0);
    int x0 = (int)x0f; x0 = x0 < 0 ? 0 : (x0 > W - 1 ? W - 1 : x0);
    const int y1 = (y0 + 1 > H - 1) ? (H - 1) : (y0 + 1);
    const int x1 = (x0 + 1 > W - 1) ? (W - 1) : (x0 + 1);

    const float w00 = (1.0f - wy) * (1.0f - wx);
    const float w01 = (1.0f - wy) * wx;
    const float w10 = wy * (1.0f - wx);
    const float w11 = wy * wx;

    // Row base pointers (all wave-uniform; rows are 2 KB, 16B-aligned).
    const size_t base = (size_t)b * (size_t)(H * W) * C;
    const v4f* __restrict__ r00 = (const v4f*)(feat + base + (size_t)(y0 * W + x0) * C);
    const v4f* __restrict__ r01 = (const v4f*)(feat + base + (size_t)(y0 * W + x1) * C);
    const v4f* __restrict__ r10 = (const v4f*)(feat + base + (size_t)(y1 * W + x0) * C);
    const v4f* __restrict__ r11 = (const v4f*)(feat + base + (size_t)(y1 * W + x1) * C);
    v4f* __restrict__ o = (v4f*)(out + (size_t)pixel * C);

    // 512 channels = 128 float4 = 32 lanes x 4 iterations. Fully unrolled:
    // 16 outstanding B128 gathers per lane, then blend, then NT stores.
    #pragma unroll
    for (int i = 0; i < ITERS; ++i) {
        const int idx = i * 32 + lane;           // consecutive lanes -> 512B span
        const v4f a = r00[idx];
        const v4f c = r01[idx];
        const v4f d = r10[idx];
        const v4f e = r11[idx];
        v4f r;
        r.x = a.x * w00 + c.x * w01 + d.x * w10 + e.x * w11;
        r.y = a.y * w00 + c.y * w01 + d.y * w10 + e.y * w11;
        r.z = a.z * w00 + c.z * w01 + d.z * w10 + e.z * w11;
        r.w = a.w * w00 + c.w * w01 + d.w * w10 + e.w * w11;
        // Output is streamed once and never re-read: non-temporal store
        // (global_store_b128 th:NT) preserves L2 for the feature map.
        __builtin_nontemporal_store(r, &o[idx]);
    }
}

extern "C" void kernel_launch(void* const* d_in, const int* in_sizes, int n_in,
                              void* d_out, int out_size, void* d_ws, size_t ws_size,
                              hipStream_t stream) {
    const float* feat = (const float*)d_in[0];   // [128,37,50,512] f32
    const float* grid = (const float*)d_in[1];   // [128,14,14,2]  f32
    float* out = (float*)d_out;                  // [128,14,14,512] f32

    constexpr int TOTAL = 128 * 14 * 14;         // 25088 pixels
    constexpr int WAVES_PER_BLOCK = 8;           // 256 threads / wave32
    const int blocks = (TOTAL + WAVES_PER_BLOCK - 1) / WAVES_PER_BLOCK;  // 3136

    _RoICrop_kernel<<<blocks, 256, 0, stream>>>(feat, grid, out);
}